// GCN_32444182954549
// MI455X (gfx1250) — compile-verified
//
#include <hip/hip_runtime.h>

#define NN 40000
#define EE 640000
#define DD 128

typedef __attribute__((ext_vector_type(2))) float v2f;
typedef __attribute__((ext_vector_type(8))) float v8f;

// ---------------- utility kernels ----------------

__global__ void gcn_fill_zero(float* __restrict__ p, int n) {
  int i = blockIdx.x * blockDim.x + threadIdx.x;
  if (i < n) p[i] = 0.0f;
}

__global__ void gcn_degree(const int* __restrict__ src, const int* __restrict__ dst,
                           float* __restrict__ deg_out, float* __restrict__ deg_in, int E) {
  int e = blockIdx.x * blockDim.x + threadIdx.x;
  if (e < E) {
    atomicAdd(&deg_out[src[e]], 1.0f);
    atomicAdd(&deg_in[dst[e]], 1.0f);
  }
}

__global__ void gcn_norm(float* __restrict__ deg_out, float* __restrict__ deg_in, int n) {
  int i = blockIdx.x * blockDim.x + threadIdx.x;
  if (i < n) {
    deg_out[i] = rsqrtf(fmaxf(deg_out[i], 1.0f));
    deg_in[i]  = rsqrtf(fmaxf(deg_in[i], 1.0f));
  }
}

// One wave (32 lanes) per edge; each lane moves 4 contiguous floats (float4 gather,
// 4 global_atomic_add_f32 scatter). Out-norm scaling fused into the gather.
__global__ __launch_bounds__(256) void gcn_scatter(
    const float* __restrict__ feat, const int* __restrict__ src, const int* __restrict__ dst,
    const float* __restrict__ outnorm, float* __restrict__ agg, int E) {
  long long tid = (long long)blockIdx.x * blockDim.x + threadIdx.x;
  int e    = (int)(tid >> 5);
  int lane = (int)(tid & 31);
  if (e >= E) return;
  int s = src[e];
  int d = dst[e];
  float sc = outnorm[s];
  float4 v = *((const float4*)(feat + (size_t)s * DD) + lane);
  float* ap = agg + (size_t)d * DD + lane * 4;
  atomicAdd(ap + 0, v.x * sc);
  atomicAdd(ap + 1, v.y * sc);
  atomicAdd(ap + 2, v.z * sc);
  atomicAdd(ap + 3, v.w * sc);
}

// ---------------- WMMA f32 GEMM: out[16 x C] tile per block ----------------
// A:[nrows x K] (optionally row-scaled), W:[K x C], bias:[C]; out = act(A@W + b).
// blockDim.x = 32 * (C/16); wave w computes the 16x16 tile at columns [16w, 16w+16).
// A-fragment layout per ISA (16x4 f32 A): v0 = A[m][k0+2*half], v1 = A[m][k0+2*half+1],
// half = lane>=16 -> contiguous float2 from LDS (ds_load_b64).
template <int K, int C, int RELU>
__global__ __launch_bounds__(32 * (C / 16)) void gcn_wmma_gemm(
    const float* __restrict__ A, const float* __restrict__ W,
    const float* __restrict__ bias, const float* __restrict__ rowscale,
    float* __restrict__ out) {
  __shared__ float Ash[16 * K];

  const int tid   = threadIdx.x;
  const int rbase = blockIdx.x * 16;

  // Stage 16 x K A-tile into LDS, fusing the per-row in-norm scale.
  for (int idx = tid; idx < 16 * K; idx += blockDim.x) {
    int r = idx / K;
    int k = idx - r * K;
    float v = A[(size_t)(rbase + r) * K + k];
    if (rowscale) v *= rowscale[rbase + r];
    Ash[idx] = v;
  }
  __syncthreads();

  const int wave  = tid >> 5;
  const int lane  = tid & 31;
  const int half  = lane >> 4;   // 0: K pair (k0,k0+1); 1: K pair (k0+2,k0+3)
  const int m     = lane & 15;
  const int cbase = wave * 16;

  v8f acc = {};
  const float* arow = &Ash[m * K];
#pragma unroll
  for (int k0 = 0; k0 < K; k0 += 4) {
    v2f a, b;
    const int ka = k0 + 2 * half;
    a.x = arow[ka];
    a.y = arow[ka + 1];
    b.x = W[(size_t)ka * C + cbase + m];
    b.y = W[(size_t)(ka + 1) * C + cbase + m];
    acc = __builtin_amdgcn_wmma_f32_16x16x4_f32(
        /*neg_a=*/false, a, /*neg_b=*/false, b,
        /*c_mod=*/(short)0, acc, /*reuse_a=*/false, /*reuse_b=*/false);
  }

  // C/D layout: VGPR r -> row (r + 8*half), col = lane&15.
  const int col = cbase + m;
  const float bv = bias[col];
#pragma unroll
  for (int r = 0; r < 8; ++r) {
    int row = rbase + r + 8 * half;
    float v = acc[r] + bv;
    if (RELU) v = fmaxf(v, 0.0f);
    out[(size_t)row * C + col] = v;
  }
}

// Final head: [N,64] @ [64,10] + b. Tiny K, tiny C -> plain VALU kernel.
__global__ __launch_bounds__(256) void gcn_head(
    const float* __restrict__ h, const float* __restrict__ Wl2,
    const float* __restrict__ bl2, float* __restrict__ out, int n) {
  int tid = blockIdx.x * blockDim.x + threadIdx.x;
  if (tid >= n * 10) return;
  int row = tid / 10;
  int c   = tid - row * 10;
  const float* hp = h + (size_t)row * 64;
  float acc = bl2[c];
#pragma unroll
  for (int k = 0; k < 64; ++k) acc = fmaf(hp[k], Wl2[k * 10 + c], acc);
  out[tid] = acc;
}

// ---------------- launch ----------------

extern "C" void kernel_launch(void* const* d_in, const int* in_sizes, int n_in,
                              void* d_out, int out_size, void* d_ws, size_t ws_size,
                              hipStream_t stream) {
  const float* h   = (const float*)d_in[0];   // [N,128]
  const int*   src = (const int*)d_in[1];     // [E]
  const int*   dst = (const int*)d_in[2];     // [E]
  const float* W0  = (const float*)d_in[3];   // [128,128]
  const float* b0  = (const float*)d_in[4];
  const float* W1  = (const float*)d_in[5];   // [128,128]
  const float* b1  = (const float*)d_in[6];
  const float* Wl1 = (const float*)d_in[7];   // [128,64]
  const float* bl1 = (const float*)d_in[8];
  const float* Wl2 = (const float*)d_in[9];   // [64,10]
  const float* bl2 = (const float*)d_in[10];
  float* out = (float*)d_out;                 // [N,10]

  float* wsf      = (float*)d_ws;
  float* deg_out  = wsf;                  // N  (becomes out-norm in place)
  float* deg_in   = wsf + NN;             // N  (becomes in-norm in place)
  float* buf1     = wsf + 2 * NN;         // N*128
  float* buf2     = buf1 + (size_t)NN * DD; // N*128

  const int TPB = 256;
  const int feat_elems = NN * DD;
  dim3 featGrid((feat_elems + TPB - 1) / TPB);
  dim3 scatGrid((EE * 32) / TPB);
  dim3 gemmGrid(NN / 16);  // 2500, exact

  // Degrees + norms
  gcn_fill_zero<<<dim3((2 * NN + TPB - 1) / TPB), TPB, 0, stream>>>(deg_out, 2 * NN);
  gcn_degree<<<dim3((EE + TPB - 1) / TPB), TPB, 0, stream>>>(src, dst, deg_out, deg_in, EE);
  gcn_norm<<<dim3((NN + TPB - 1) / TPB), TPB, 0, stream>>>(deg_out, deg_in, NN);

  // Layer 0: scatter(h * out-norm) -> buf2; relu((in-norm * buf2) @ W0 + b0) -> buf1
  gcn_fill_zero<<<featGrid, TPB, 0, stream>>>(buf2, feat_elems);
  gcn_scatter<<<scatGrid, TPB, 0, stream>>>(h, src, dst, deg_out, buf2, EE);
  gcn_wmma_gemm<128, 128, 1><<<gemmGrid, 256, 0, stream>>>(buf2, W0, b0, deg_in, buf1);

  // Layer 1
  gcn_fill_zero<<<featGrid, TPB, 0, stream>>>(buf2, feat_elems);
  gcn_scatter<<<scatGrid, TPB, 0, stream>>>(buf1, src, dst, deg_out, buf2, EE);
  gcn_wmma_gemm<128, 128, 1><<<gemmGrid, 256, 0, stream>>>(buf2, W1, b1, deg_in, buf1);

  // linear1: relu(buf1 @ Wl1 + bl1) -> buf2 [N,64]
  gcn_wmma_gemm<128, 64, 1><<<gemmGrid, 128, 0, stream>>>(buf1, Wl1, bl1, nullptr, buf2);

  // linear2 head -> out [N,10]
  gcn_head<<<dim3((NN * 10 + TPB - 1) / TPB), TPB, 0, stream>>>(buf2, Wl2, bl2, out, NN);
}